// ArithmeticGreyboxModule_20220706030182
// MI455X (gfx1250) — compile-verified
//
#include <hip/hip_runtime.h>
#include <math.h>

// ---------------------------------------------------------------------------
// ArithmeticGreyboxModule: out = (1-sigmoid(blend))*x + sigmoid(blend)*sym
// where sym differs from x only in a tiny token-injection region (t < 20).
// Memory-bound streaming op: 541 MB traffic -> ~23 us floor at 23.3 TB/s.
// Strategy: NT b128 streaming bulk kernel + tiny exact fixup kernel.
// ---------------------------------------------------------------------------

typedef float f32x4 __attribute__((ext_vector_type(4)));

__device__ __forceinline__ float device_sigmoid(float s) {
    return 1.0f / (1.0f + expf(-s));
}

// Bulk kernel: every element gets (1-b)*x + b*x (reference applies the blend
// expression everywhere; where sym == x this is NOT bit-identical to x, so we
// must actually compute it). 128-bit non-temporal loads/stores, prefetch one
// grid stride ahead.
__global__ void __launch_bounds__(256)
greybox_blend_bulk(const float* __restrict__ x,
                   const float* __restrict__ blend_p,
                   float* __restrict__ out,
                   long long n4, long long n)
{
    const float b  = device_sigmoid(blend_p[0]);   // uniform -> scalar load
    const float ob = 1.0f - b;

    const f32x4* __restrict__ xv = (const f32x4*)x;
    f32x4* __restrict__ ov       = (f32x4*)out;

    const long long stride = (long long)gridDim.x * blockDim.x;
    long long i = (long long)blockIdx.x * blockDim.x + threadIdx.x;

    for (; i < n4; i += stride) {
        // gfx1250 prefetch (global_prefetch_b8); speculative, OOB is dropped.
        __builtin_prefetch(&xv[i + stride], 0, 0);
        f32x4 v = __builtin_nontemporal_load(&xv[i]);
        f32x4 r = ob * v + b * v;                  // matches (1-b)*x + b*sym
        __builtin_nontemporal_store(r, &ov[i]);
    }

    // Scalar tail (n not divisible by 4); for this problem n % 4 == 0.
    long long t = n4 * 4 + (long long)blockIdx.x * blockDim.x + threadIdx.x;
    for (; t < n; t += stride) {
        float v = x[t];
        out[t] = ob * v + b * v;
    }
}

// Fixup kernel: exact _inject() semantics for the region t in [0, 20).
// Covers every token case (START/digit/PLUS/MINUS/EQUALS) read from device,
// then rewrites out = (1-b)*x + b*sym there. 8*20*258 = 41,280 threads.
__global__ void __launch_bounds__(256)
greybox_inject_fixup(const float* __restrict__ x,
                     const float* __restrict__ blend_p,
                     const int* __restrict__ src_p,
                     float* __restrict__ out,
                     int B, int T_, int C /* = 2*FD */)
{
    const int idx   = blockIdx.x * blockDim.x + threadIdx.x;
    const int total = B * 20 * C;
    if (idx >= total) return;

    const int c    = idx % C;          // channel within row: c = fd*2 + comp
    const int tmp  = idx / C;
    const int t    = tmp % 20;         // sequence position (only t<20 matter)
    const int bi   = tmp / 20;         // batch
    const int fd   = c >> 1;           // frequency bin
    const int comp = c & 1;            // 0 = real, 1 = imag

    const long long off = ((long long)bi * T_ + t) * (long long)C + c;
    const float xv = x[off];
    float symv = xv;

    const int src = src_p[0];

    // START (token 0): zero the whole protected block (t < 20, all fd).
    if (src == 0) {
        symv = 0.0f;
    }
    // Digit tokens 1..10: clear digit registers at fd==0, set one-hot digit.
    if (src >= 1 && src <= 10) {
        if (fd == 0) {
            if (t >= 2 && t <= 11) symv = 0.0f;
            const int digit_val = (src - 1) % 10;
            if (t == 2 + digit_val) symv = (comp == 0) ? 1.0f : 0.0f;
        }
    }
    // PLUS (11): op register (t==1, fd==0) = (1, 0)
    if (src == 11) {
        if (fd == 0 && t == 1) symv = (comp == 0) ? 1.0f : 0.0f;
    }
    // MINUS (12): op register = (-1, 0)
    if (src == 12) {
        if (fd == 0 && t == 1) symv = (comp == 0) ? -1.0f : 0.0f;
    }
    // EQUALS (13): zero result/op/digit registers at fd==0 (initial state).
    if (src == 13) {
        if (fd == 0 &&
            (t == 14 || t == 15 || t == 16 || t == 1 || (t >= 2 && t <= 11)))
            symv = 0.0f;
    }

    const float b = device_sigmoid(blend_p[0]);
    out[off] = (1.0f - b) * xv + b * symv;
}

extern "C" void kernel_launch(void* const* d_in, const int* in_sizes, int n_in,
                              void* d_out, int out_size, void* d_ws, size_t ws_size,
                              hipStream_t stream) {
    (void)n_in; (void)d_ws; (void)ws_size; (void)out_size;

    const float* x       = (const float*)d_in[0];  // carrier_freq_flat, f32
    const float* blend_p = (const float*)d_in[1];  // symbolic_blend scalar, f32
    const int*   src_p   = (const int*)d_in[2];    // src_token scalar, int
    // d_in[3] (tgt_token) is unused by the reference computation.
    float* out = (float*)d_out;

    const long long n  = (long long)in_sizes[0];   // 8 * 32768 * 258
    const long long n4 = n / 4;

    const int C  = 258;                            // 2 * FD, FD = 129
    const int T_ = 32768;                          // sequence length
    const int B  = (int)(n / ((long long)T_ * (long long)C));

    // Bulk: 4 float4 per thread via grid-stride; enough blocks to fill the
    // device while amortizing address setup.
    const int threads = 256;
    long long want = (n4 + (long long)threads * 4 - 1) / ((long long)threads * 4);
    if (want < 1) want = 1;
    if (want > 1048576) want = 1048576;
    greybox_blend_bulk<<<(int)want, threads, 0, stream>>>(x, blend_p, out, n4, n);

    // Fixup: exact injection semantics over the t<20 region (runs after bulk
    // in stream order, overwriting only the affected elements).
    const int total  = B * 20 * C;
    const int fblk   = (total + 255) / 256;
    greybox_inject_fixup<<<fblk, 256, 0, stream>>>(x, blend_p, src_p, out, B, T_, C);
}